// PMField_45569603010669
// MI455X (gfx1250) — compile-verified
//
#include <hip/hip_runtime.h>

typedef float v2f __attribute__((ext_vector_type(2)));
typedef float v8f __attribute__((ext_vector_type(8)));
typedef int   v4i_g __attribute__((vector_size(16)));  // matches builtin pointee

#define NB 131072
#define NC 256
#define ND 64
#define NSTEPS 8
#define DTBETA 0.15f
#define EPS_ 1e-4f

#define CSTR 68   // padded LDS row stride for centers (272B: 16B aligned, conflict-free frags)
#define ZSTR 68   // per-wave z staging stride [M][K]
#define WSTR 20   // per-wave w staging stride [M][Kc]
#define WAVES 8
#define RPW 16                      // rows per wave (one 16xD tile)
#define RPB (WAVES * RPW)           // 128 rows per block

// ---- CDNA5 async global->LDS path (ASYNCcnt-tracked), with sync fallback ----
#if defined(__has_builtin)
#if __has_builtin(__builtin_amdgcn_global_load_async_to_lds_b128) && \
    __has_builtin(__builtin_amdgcn_s_wait_asynccnt)
#define HAVE_ASYNC_LDS 1
#endif
#endif
#ifndef HAVE_ASYNC_LDS
#define HAVE_ASYNC_LDS 0
#endif

#if HAVE_ASYNC_LDS
typedef __attribute__((address_space(1))) v4i_g glb_v4i;  // prints as __device__
typedef __attribute__((address_space(3))) v4i_g lds_v4i;  // prints as __shared__
static __device__ inline glb_v4i* to_glb(const void* p) {
  return (glb_v4i*)(unsigned long long)p;  // integer -> AS1 pointer
}
static __device__ inline lds_v4i* to_lds(void* p) {
  // generic LDS address: low 32 bits are the LDS byte offset (aperture rule)
  return (lds_v4i*)(unsigned int)(unsigned long long)p;
}
#endif

static __device__ inline v8f wmma4(v2f a, v2f b, v8f c) {
  // V_WMMA_F32_16X16X4_F32 : D = A(16x4 f32) * B(4x16 f32) + C(16x16 f32)
  return __builtin_amdgcn_wmma_f32_16x16x4_f32(false, a, false, b, (short)0, c,
                                               false, false);
}

__global__ __launch_bounds__(256) void pm_field_kernel(
    const float* __restrict__ zin, const float* __restrict__ centers,
    const float* __restrict__ mus, float* __restrict__ out) {
  __shared__ float sC[NC * CSTR];          // centers, padded row stride
  __shared__ float sCn2[NC];               // |c|^2
  __shared__ float sMu[NC];
  __shared__ float sZ[WAVES][RPW * ZSTR];  // per-wave z transpose staging
  __shared__ float sW[WAVES][RPW * WSTR];  // per-wave w transpose staging

  const int tid  = threadIdx.x;
  const int lane = tid & 31;
  const int wave = tid >> 5;
  const int lm   = lane & 15;   // lane within half-wave
  const int hl   = lane >> 4;   // which half-wave (0/1)

  // ---- centers -> LDS (one 64-float row per thread) + cn2 + mu ----
#if HAVE_ASYNC_LDS
  {
    const float* src = centers + tid * ND;
    float* dst = sC + tid * CSTR;
#pragma unroll
    for (int k = 0; k < ND; k += 4)
      __builtin_amdgcn_global_load_async_to_lds_b128(to_glb(src + k),
                                                     to_lds(dst + k), 0, 0);
    __builtin_amdgcn_s_wait_asynccnt(0);
  }
  __syncthreads();
  {
    const float* row = sC + tid * CSTR;
    float s = 0.f;
#pragma unroll
    for (int k = 0; k < ND; k += 4) {
      float4 v = *(const float4*)(row + k);
      s += v.x * v.x + v.y * v.y + v.z * v.z + v.w * v.w;
    }
    sCn2[tid] = s;
    sMu[tid]  = mus[tid];
  }
#else
  {
    const float* src = centers + tid * ND;
    float* dst = sC + tid * CSTR;
    float s = 0.f;
#pragma unroll
    for (int k = 0; k < ND; k += 4) {
      float4 v = *(const float4*)(src + k);
      *(float4*)(dst + k) = v;
      s += v.x * v.x + v.y * v.y + v.z * v.z + v.w * v.w;
    }
    sCn2[tid] = s;
    sMu[tid]  = mus[tid];
  }
#endif
  __syncthreads();

  const int rowBase = blockIdx.x * RPB + wave * RPW;

  // z tile in C/D layout: zc[ch][v] = z[rowBase + v + 8*hl][ch*16 + lm]
  v8f zc[4];
#pragma unroll
  for (int ch = 0; ch < 4; ++ch)
#pragma unroll
    for (int v = 0; v < 8; ++v)
      zc[ch][v] = zin[(rowBase + v + 8 * hl) * ND + ch * 16 + lm];

  float* zs = &sZ[wave][0];
  float* ws = &sW[wave][0];
  const v8f vzero = {0.f, 0.f, 0.f, 0.f, 0.f, 0.f, 0.f, 0.f};

  for (int step = 0; step < NSTEPS; ++step) {
    // ---- stage z (C-layout -> LDS [M][K]) and pull A-layout fragments ----
#pragma unroll
    for (int ch = 0; ch < 4; ++ch)
#pragma unroll
      for (int v = 0; v < 8; ++v)
        zs[(v + 8 * hl) * ZSTR + ch * 16 + lm] = zc[ch][v];

    v2f zA[16];  // A frag k: lane holds (M=lm, K=4k+2*hl .. +1)
#pragma unroll
    for (int k = 0; k < 16; ++k)
      zA[k] = *(const v2f*)(zs + lm * ZSTR + 4 * k + 2 * hl);

    // ---- |z|^2 per row, broadcast in C-layout via half-wave xor-reduce ----
    float zz[8];
#pragma unroll
    for (int v = 0; v < 8; ++v) {
      float s = 0.f;
#pragma unroll
      for (int ch = 0; ch < 4; ++ch) s += zc[ch][v] * zc[ch][v];
      zz[v] = s;
    }
#pragma unroll
    for (int m = 1; m <= 8; m <<= 1)
#pragma unroll
      for (int v = 0; v < 8; ++v) zz[v] += __shfl_xor(zz[v], m, 32);

    v8f g[4];
#pragma unroll
    for (int ch = 0; ch < 4; ++ch) g[ch] = vzero;
    float an[8], sw[8];
#pragma unroll
    for (int v = 0; v < 8; ++v) { an[v] = 0.f; sw[v] = 0.f; }

    // ---- fused loop over 16-center tiles ----
    for (int ct = 0; ct < 16; ++ct) {
      // GEMM1: S[16x16] = z_tile @ centers_tile^T (16 WMMAs, K=64),
      // two independent accumulator chains for ILP
      v8f S0 = vzero, S1 = vzero;
      const float* cb = sC + (ct * 16 + lm) * CSTR + 2 * hl;
#pragma unroll
      for (int k = 0; k < 16; k += 2) {
        v2f b0 = *(const v2f*)(cb + 4 * k);
        v2f b1 = *(const v2f*)(cb + 4 * k + 4);
        S0 = wmma4(zA[k], b0, S0);
        S1 = wmma4(zA[k + 1], b1, S1);
      }
      v8f S = S0 + S1;

      const float cn2l = sCn2[ct * 16 + lm];
      const float mul  = sMu[ct * 16 + lm];
      float wv[8];
#pragma unroll
      for (int v = 0; v < 8; ++v) {
        float r2 = fmaxf(zz[v] + cn2l - 2.0f * S[v], 0.0f) + EPS_;
        float ri = rsqrtf(r2);          // 1/r
        an[v] += mul * ri;              // mu / r
        float wq = mul * ri * ri * ri;  // mu / r^3
        sw[v] += wq;
        wv[v] = wq;
      }

      // stage w tile (C-layout -> LDS [M][Kc]) and pull A-layout fragments
#pragma unroll
      for (int v = 0; v < 8; ++v)
        ws[(v + 8 * hl) * WSTR + lm] = wv[v];
      v2f wA[4];
#pragma unroll
      for (int j = 0; j < 4; ++j)
        wA[j] = *(const v2f*)(ws + lm * WSTR + 4 * j + 2 * hl);

      // GEMM2: g[16x64] += w_tile @ centers_tile (16 WMMAs, K=16,
      // 4 independent accumulator chains across the N-chunks)
#pragma unroll
      for (int j2 = 0; j2 < 4; ++j2) {
        const float* bp = sC + (ct * 16 + 4 * j2 + 2 * hl) * CSTR;
#pragma unroll
        for (int nj = 0; nj < 4; ++nj) {
          v2f b2;
          b2[0] = bp[nj * 16 + lm];
          b2[1] = bp[CSTR + nj * 16 + lm];
          g[nj] = wmma4(wA[j2], b2, g[nj]);
        }
      }
    }

    // ---- row reductions for n = 1 + sum(mu/r) and sum(w) ----
#pragma unroll
    for (int m = 1; m <= 8; m <<= 1)
#pragma unroll
      for (int v = 0; v < 8; ++v) {
        an[v] += __shfl_xor(an[v], m, 32);
        sw[v] += __shfl_xor(sw[v], m, 32);
      }

    // ---- update z: z = clip(z + dt*beta*(g - z*sw)/n, -3, 3) ----
#pragma unroll
    for (int v = 0; v < 8; ++v) {
      float ninv = 1.0f / (1.0f + an[v]);
#pragma unroll
      for (int ch = 0; ch < 4; ++ch) {
        float gg = (g[ch][v] - zc[ch][v] * sw[v]) * ninv;
        float zn = zc[ch][v] + DTBETA * gg;
        zc[ch][v] = fminf(fmaxf(zn, -3.0f), 3.0f);
      }
    }
  }

  // ---- write back final z ----
#pragma unroll
  for (int ch = 0; ch < 4; ++ch)
#pragma unroll
    for (int v = 0; v < 8; ++v)
      out[(rowBase + v + 8 * hl) * ND + ch * 16 + lm] = zc[ch][v];
}

extern "C" void kernel_launch(void* const* d_in, const int* in_sizes, int n_in,
                              void* d_out, int out_size, void* d_ws,
                              size_t ws_size, hipStream_t stream) {
  (void)in_sizes; (void)n_in; (void)d_ws; (void)ws_size; (void)out_size;
  const float* z       = (const float*)d_in[0];
  const float* centers = (const float*)d_in[1];
  const float* mus     = (const float*)d_in[2];
  float* out           = (float*)d_out;
  dim3 grid(NB / RPB), block(256);
  pm_field_kernel<<<grid, block, 0, stream>>>(z, centers, mus, out);
}